// LSTMArgSelector_19997367730783
// MI455X (gfx1250) — compile-verified
//
#include <hip/hip_runtime.h>
#include <hip/hip_bf16.h>

typedef float v2f __attribute__((ext_vector_type(2)));
typedef float v8f __attribute__((ext_vector_type(8)));

#define Hh 512
#define Ee 512
#define Bb 256
#define Tt 64
#define G4 2048   // 4*H
#define LDSW 516  // padded LDS row stride (conflict-free: bank = 4m + ka)

__device__ __forceinline__ float sigf(float x) {
    return 1.0f / (1.0f + __expf(-x));
}

// ---------------------------------------------------------------------------
// init: hbuf[0]=init_state, cbuf[0]=init_state, state[t=0]=init_state, out=0
// ---------------------------------------------------------------------------
__global__ __launch_bounds__(256) void init_kernel(const float* __restrict__ init_state,
                                                   float* __restrict__ hbuf,
                                                   float* __restrict__ cbuf,
                                                   float* __restrict__ state,
                                                   float* __restrict__ out) {
    int idx = blockIdx.x * 256 + threadIdx.x;   // 512 blocks -> 131072 = B*H
    float v = init_state[idx];
    hbuf[idx]  = v;
    cbuf[idx]  = v;
    state[idx] = v;                              // state[0, b, j]
    if (idx < Bb) out[idx] = 0.0f;
}

// ---------------------------------------------------------------------------
// Xg = gather(choice_embed, arg_seq) @ Wx + b   -> [T*B, 4H], row r = t*B + b
// One wave per 16x16 output tile; fp32 WMMA, K stepped by 4.
// ---------------------------------------------------------------------------
__global__ __launch_bounds__(256) void xg_gemm(const float* __restrict__ emb,
                                               const int*   __restrict__ seq,
                                               const float* __restrict__ Wx,
                                               const float* __restrict__ bias,
                                               float*       __restrict__ Xg) {
    const int lane = threadIdx.x & 31;
    const int wave = (blockIdx.x * 256 + threadIdx.x) >> 5;  // 0..131071
    const int tr = wave >> 7;        // row tile 0..1023  (T*B/16)
    const int tn = wave & 127;       // col tile 0..127   (4H/16)
    const int rows0 = tr * 16;
    const int m  = lane & 15;
    const int hi = lane >> 4;
    const int n  = tn * 16 + m;

    const int row = rows0 + m;       // r = t*B + b
    const int t = row >> 8;
    const int b = row & 255;
    const int eidx = seq[b * Tt + t];
    const float* __restrict__ arow = emb + (size_t)eidx * Ee;

    v8f acc;
    const float bv = bias[n];
    #pragma unroll
    for (int r = 0; r < 8; ++r) acc[r] = bv;

    for (int k0 = 0; k0 < Ee; k0 += 4) {
        const int ka = k0 + 2 * hi;
        v2f a = *(const v2f*)(arow + ka);
        v2f bm;
        bm[0] = Wx[(size_t)ka * G4 + n];
        bm[1] = Wx[(size_t)(ka + 1) * G4 + n];
        acc = __builtin_amdgcn_wmma_f32_16x16x4_f32(false, a, false, bm,
                                                    (short)0, acc, false, false);
    }

    #pragma unroll
    for (int r = 0; r < 8; ++r) {
        const int rr = rows0 + r + 8 * hi;
        Xg[(size_t)rr * G4 + n] = acc[r];
    }
}

// ---------------------------------------------------------------------------
// One LSTM step: gates = Xg[t] + h@Wh ; fused cell update.
// Block = 8 waves, one batch-tile (16 rows of h staged once into LDS via
// async-to-LDS), 4 hidden-unit tiles, 2 waves per tile splitting K=512.
// 1024 waves -> 128 blocks of 256.
// ---------------------------------------------------------------------------
__global__ __launch_bounds__(256) void lstm_step(const float* __restrict__ Xg,
                                                 const float* __restrict__ Wh,
                                                 const float* __restrict__ hin,
                                                 const float* __restrict__ cin,
                                                 float*       __restrict__ hout,
                                                 float*       __restrict__ cout,
                                                 float*       __restrict__ state,
                                                 int t) {
    __shared__ float lds_h[16 * LDSW];          // padded 16x512 h tile (~33KB)
    __shared__ float lds_p[4 * 4 * 32 * 8];     // [jt_local][gate][lane][r] (16KB)

    const int tid  = threadIdx.x;
    const int lane = tid & 31;
    const int wid  = tid >> 5;                  // 0..7
    const int bt = blockIdx.x >> 3;             // batch tile 0..15
    const int b0 = bt * 16;
    const int jt_local = wid >> 1;              // 0..3
    const int kh = wid & 1;                     // K-half
    const int jt = (blockIdx.x & 7) * 4 + jt_local;  // 0..31
    const int m  = lane & 15;
    const int hi = lane >> 4;
    const int j  = jt * 16 + m;

    // ---- stage the shared 16x512 h tile into padded LDS (async, b128) ----
    #pragma unroll
    for (int i = 0; i < 8; ++i) {
        const int fid = tid + 256 * i;          // 0..2047 float4s
        const int row = fid >> 7;               // 0..15
        const int c4  = (fid & 127) << 2;       // float col
        const uint32_t loff = (uint32_t)(uintptr_t)&lds_h[row * LDSW + c4];
        const float* g = hin + (size_t)(b0 + row) * Hh + c4;
        asm volatile("global_load_async_to_lds_b128 %0, %1, off"
                     :: "v"(loff), "v"(g) : "memory");
    }
    asm volatile("s_wait_asynccnt 0" ::: "memory");
    __syncthreads();

    v8f aI, aF, aG, aO;
    if (kh == 0) {
        // kh=0 seeds from Xg (already contains x@Wx + b)
        const float* __restrict__ xg = Xg + (size_t)t * Bb * G4;
        #pragma unroll
        for (int r = 0; r < 8; ++r) {
            const size_t ro = (size_t)(b0 + r + 8 * hi) * G4;
            aI[r] = xg[ro + j];
            aF[r] = xg[ro + 512 + j];
            aG[r] = xg[ro + 1024 + j];
            aO[r] = xg[ro + 1536 + j];
        }
    } else {
        #pragma unroll
        for (int r = 0; r < 8; ++r) { aI[r] = 0.f; aF[r] = 0.f; aG[r] = 0.f; aO[r] = 0.f; }
    }

    const float* __restrict__ lrow = &lds_h[m * LDSW];
    const int kbeg = kh * 256;
    for (int k0 = kbeg; k0 < kbeg + 256; k0 += 4) {
        const int ka = k0 + 2 * hi;
        v2f a = *(const v2f*)(lrow + ka);       // ds_load_b64, conflict-free
        const float* __restrict__ w0 = Wh + (size_t)ka * G4;
        const float* __restrict__ w1 = Wh + (size_t)(ka + 1) * G4;
        v2f bI; bI[0] = w0[j];        bI[1] = w1[j];
        v2f bF; bF[0] = w0[512 + j];  bF[1] = w1[512 + j];
        v2f bG; bG[0] = w0[1024 + j]; bG[1] = w1[1024 + j];
        v2f bO; bO[0] = w0[1536 + j]; bO[1] = w1[1536 + j];
        aI = __builtin_amdgcn_wmma_f32_16x16x4_f32(false, a, false, bI, (short)0, aI, false, false);
        aF = __builtin_amdgcn_wmma_f32_16x16x4_f32(false, a, false, bF, (short)0, aF, false, false);
        aG = __builtin_amdgcn_wmma_f32_16x16x4_f32(false, a, false, bG, (short)0, aG, false, false);
        aO = __builtin_amdgcn_wmma_f32_16x16x4_f32(false, a, false, bO, (short)0, aO, false, false);
    }

    // ---- combine K-halves through LDS ----
    if (kh == 1) {
        const int base = ((jt_local * 4) * 32 + lane) * 8;
        *((v8f*)&lds_p[base          ]) = aI;
        *((v8f*)&lds_p[base + 32 * 8 ]) = aF;
        *((v8f*)&lds_p[base + 64 * 8 ]) = aG;
        *((v8f*)&lds_p[base + 96 * 8 ]) = aO;
    }
    __syncthreads();

    if (kh == 0) {
        const int base = ((jt_local * 4) * 32 + lane) * 8;
        aI += *((const v8f*)&lds_p[base          ]);
        aF += *((const v8f*)&lds_p[base + 32 * 8 ]);
        aG += *((const v8f*)&lds_p[base + 64 * 8 ]);
        aO += *((const v8f*)&lds_p[base + 96 * 8 ]);

        // fused cell update; each (r, lane) is a unique (batch, unit)
        #pragma unroll
        for (int r = 0; r < 8; ++r) {
            const int bbx = b0 + r + 8 * hi;
            const size_t off = (size_t)bbx * Hh + j;
            const float iv = sigf(aI[r]);
            const float fv = sigf(aF[r]);
            const float gv = tanhf(aG[r]);
            const float ov = sigf(aO[r]);
            const float c2 = fv * cin[off] + iv * gv;
            const float h2 = ov * tanhf(c2);
            cout[off] = c2;
            hout[off] = h2;
            if (t + 1 < Tt) state[(size_t)(t + 1) * Bb * Hh + off] = h2;
        }
    }
}

// ---------------------------------------------------------------------------
// Head: relu(concat(state, x) @ W1 + b1) @ W2 + b2, summed over T into out[b].
// Fused epilogue: ReLU + W2 dot + half-wave reduce + atomicAdd.
// 16384 waves -> 2048 blocks of 256.
// ---------------------------------------------------------------------------
__global__ __launch_bounds__(256) void head_kernel(const float* __restrict__ state,
                                                   const float* __restrict__ emb,
                                                   const int*   __restrict__ seq,
                                                   const float* __restrict__ W1,
                                                   const float* __restrict__ b1,
                                                   const float* __restrict__ W2,
                                                   const float* __restrict__ b2,
                                                   float*       __restrict__ out) {
    const int lane = threadIdx.x & 31;
    const int wave = (blockIdx.x * 256 + threadIdx.x) >> 5;  // 0..16383
    const int tr = wave >> 4;        // row tile 0..1023 over r = t*B + b
    const int tn = wave & 15;        // col tile 0..15  (256/16)
    const int rows0 = tr * 16;
    const int t  = rows0 >> 8;
    const int b0 = rows0 & 255;
    const int m  = lane & 15;
    const int hi = lane >> 4;
    const int n  = tn * 16 + m;

    const int eidx = seq[(b0 + m) * Tt + t];
    const float* __restrict__ srow = state + ((size_t)t * Bb + b0 + m) * Hh;
    const float* __restrict__ erow = emb + (size_t)eidx * Ee;

    v8f acc;
    #pragma unroll
    for (int r = 0; r < 8; ++r) acc[r] = 0.0f;

    for (int k0 = 0; k0 < Hh + Ee; k0 += 4) {
        const int ka = k0 + 2 * hi;               // K-chunks never straddle 512
        v2f a = (ka < Hh) ? *(const v2f*)(srow + ka)
                          : *(const v2f*)(erow + (ka - Hh));
        v2f bm;
        bm[0] = W1[(size_t)ka * 256 + n];
        bm[1] = W1[(size_t)(ka + 1) * 256 + n];
        acc = __builtin_amdgcn_wmma_f32_16x16x4_f32(false, a, false, bm,
                                                    (short)0, acc, false, false);
    }

    const float b1n = b1[n];
    const float w2n = W2[n];
    #pragma unroll
    for (int r = 0; r < 8; ++r) {
        float hv = acc[r] + b1n;
        hv = hv > 0.0f ? hv : 0.0f;
        float s = hv * w2n;
        // reduce over the 16-lane half: lanes 0-15 hold row r, 16-31 hold row r+8
        #pragma unroll
        for (int off = 8; off > 0; off >>= 1) s += __shfl_xor(s, off, 16);
        if (m == 0) {
            if (tn == 0) s += b2[0];              // b2 once per (b, t)
            atomicAdd(out + b0 + r + 8 * hi, s);
        }
    }
}

// ---------------------------------------------------------------------------
extern "C" void kernel_launch(void* const* d_in, const int* in_sizes, int n_in,
                              void* d_out, int out_size, void* d_ws, size_t ws_size,
                              hipStream_t stream) {
    const float* init_state = (const float*)d_in[0];   // [B,H]
    const float* emb        = (const float*)d_in[1];   // [V,E]
    const int*   seq        = (const int*)  d_in[2];   // [B,T]
    const float* Wx         = (const float*)d_in[3];   // [E,4H]
    const float* Wh         = (const float*)d_in[4];   // [H,4H]
    const float* bias       = (const float*)d_in[5];   // [4H]
    const float* W1         = (const float*)d_in[6];   // [H+E,256]
    const float* b1         = (const float*)d_in[7];   // [256]
    const float* W2         = (const float*)d_in[8];   // [256,1]
    const float* b2         = (const float*)d_in[9];   // [1]
    float* out = (float*)d_out;                        // [B,1]

    float* Xg    = (float*)d_ws;                               // [T*B, 4H]  128MB
    float* state = Xg    + (size_t)Tt * Bb * G4;               // [T, B, H]   32MB
    float* hbuf  = state + (size_t)Tt * Bb * Hh;               // 2 x [B,H]    1MB
    float* cbuf  = hbuf  + (size_t)2 * Bb * Hh;                // 2 x [B,H]    1MB

    init_kernel<<<512, 256, 0, stream>>>(init_state, hbuf, cbuf, state, out);
    xg_gemm<<<16384, 256, 0, stream>>>(emb, seq, Wx, bias, Xg);

    for (int t = 0; t < Tt; ++t) {
        const float* hin = hbuf + (size_t)(t & 1) * Bb * Hh;
        const float* cin = cbuf + (size_t)(t & 1) * Bb * Hh;
        float* hout = hbuf + (size_t)((t + 1) & 1) * Bb * Hh;
        float* cout = cbuf + (size_t)((t + 1) & 1) * Bb * Hh;
        lstm_step<<<128, 256, 0, stream>>>(Xg, Wh, hin, cin, hout, cout, state, t);
    }

    head_kernel<<<2048, 256, 0, stream>>>(state, emb, seq, W1, b1, W2, b2, out);
}